// RoomImpulseResponse_24713241821624
// MI455X (gfx1250) — compile-verified
//
#include <hip/hip_runtime.h>
#include <hip/hip_bf16.h>

typedef __attribute__((ext_vector_type(2))) float v2f;
typedef __attribute__((ext_vector_type(8))) float v8f;

// 8-byte load with only 4-byte alignment guarantee (B operand stream)
typedef struct __attribute__((packed, aligned(4))) { float x, y; } f2u;

#define T_LEN   131072
#define OUT_LEN (2 * T_LEN - 1)              // 262143
#define TILE_W  512                          // outputs per block (2 subtiles of 256)
#define NTILES  ((OUT_LEN + TILE_W - 1) / TILE_W)   // 512
#define NWAVES  8

// out[c][n0 + 256*s + 16m + q] = sum_k rir[k-16m] * ap[n0 + 256*s + q + k],  s in {0,1}
//   ap[i] = audio[c][i-(T-1)] for i in [T-1, 2T-2], else 0
// A[m][k] = rir[k-16m] shared by both subtiles; B_s[k][q] = ap[n0+256s+q+k]
// Fragment layout (wave32, f32 16x16x4): lane L, h=L>>4, mq=L&15:
//   A vgpr v holds A[mq][2h+v],  B vgpr v holds B[2h+v][mq]

__device__ __forceinline__ v8f wmma4(v2f a, v2f b, v8f c) {
    return __builtin_amdgcn_wmma_f32_16x16x4_f32(false, a, false, b,
                                                 (short)0, c, false, false);
}

// Edge: clamped unconditional loads + value selects (branchless)
__device__ __forceinline__ void step_checked2(const float* __restrict__ rir,
                                              const float* __restrict__ a_ch,
                                              int k0, int rbase, int bbase,
                                              v8f& accA, v8f& accB)
{
    int r0 = k0 + rbase, r1 = r0 + 1;
    int i0 = k0 + bbase, i1 = i0 + 1;
    int j0 = i0 + 256,   j1 = i0 + 257;
    float ax  = rir[min(max(r0, 0), T_LEN - 1)];
    float ay  = rir[min(max(r1, 0), T_LEN - 1)];
    float b0x = a_ch[min(max(i0, 0), T_LEN - 1)];
    float b0y = a_ch[min(max(i1, 0), T_LEN - 1)];
    float b1x = a_ch[min(max(j0, 0), T_LEN - 1)];
    float b1y = a_ch[min(max(j1, 0), T_LEN - 1)];
    v2f a, b0, b1;
    a.x  = ((unsigned)r0 < (unsigned)T_LEN) ? ax  : 0.f;
    a.y  = ((unsigned)r1 < (unsigned)T_LEN) ? ay  : 0.f;
    b0.x = ((unsigned)i0 < (unsigned)T_LEN) ? b0x : 0.f;
    b0.y = ((unsigned)i1 < (unsigned)T_LEN) ? b0y : 0.f;
    b1.x = ((unsigned)j0 < (unsigned)T_LEN) ? b1x : 0.f;
    b1.y = ((unsigned)j1 < (unsigned)T_LEN) ? b1y : 0.f;
    accA = wmma4(a, b0, accA);
    accB = wmma4(a, b1, accB);
}

__global__ __launch_bounds__(256)
void RoomImpulseResponse_wmma_conv(const float* __restrict__ audio,
                                   const float* __restrict__ rir,
                                   float* __restrict__ out)
{
    __shared__ float red[NWAVES * TILE_W];   // 16 KB

    const int tile = blockIdx.x;
    const int ch   = blockIdx.y;
    const int n0   = tile * TILE_W;
    const int tid  = threadIdx.x;
    const int wave = tid >> 5;
    const int lane = tid & 31;
    const int h    = lane >> 4;
    const int mq   = lane & 15;

    const float* __restrict__ a_ch = audio + ch * T_LEN;

    // Nonzero K band (union over both subtiles)
    int klo = T_LEN - 272 - n0;  if (klo < 0) klo = 0;
    int khi = 2 * T_LEN - 2 - n0;
    if (khi > T_LEN - 1 + 240) khi = T_LEN - 1 + 240;

    // Per-lane constant address components
    const int rbase = 2 * h - 16 * mq;                 // into rir (even)
    const int bbase = n0 + mq + 2 * h - (T_LEN - 1);   // into audio (ap shifted)

    // Contiguous per-wave chunk of 4-wide K steps
    const int kbeg   = klo & ~3;
    const int nsteps = ((khi - kbeg) >> 2) + 1;
    const int per    = (nsteps + NWAVES - 1) / NWAVES;
    int wLo = kbeg + (wave * per) * 4;
    int wHi = kbeg + min(nsteps, (wave + 1) * per) * 4;   // exclusive

    // Interior: both subtiles fully in-range for k0 in [iLo, iHiE)
    int iLo = T_LEN - 1 - n0;  if (iLo < 240) iLo = 240;
    iLo = (iLo + 3) & ~3;
    int iHi = 2 * T_LEN - 276 - n0;  if (iHi > T_LEN - 4) iHi = T_LEN - 4;
    int iHiE = (iHi & ~3) + 4;
    if (iHiE < iLo) iHiE = iLo;

    const int mLo = min(max(wLo, iLo), wHi);
    const int mHi = max(min(wHi, iHiE), mLo);

    v8f acc0a = {0.f,0.f,0.f,0.f,0.f,0.f,0.f,0.f};
    v8f acc0b = {0.f,0.f,0.f,0.f,0.f,0.f,0.f,0.f};
    v8f acc1a = {0.f,0.f,0.f,0.f,0.f,0.f,0.f,0.f};
    v8f acc1b = {0.f,0.f,0.f,0.f,0.f,0.f,0.f,0.f};

    // Phase 1: checked prologue
    for (int k0 = wLo; k0 < mLo; k0 += 4)
        step_checked2(rir, a_ch, k0, rbase, bbase, acc0a, acc1a);

    // Phase 2: unchecked interior. Per-lane byte pointers hoisted out of the
    // loop; all 6 loads per iteration are immediate-offset b64 loads; A is
    // reused by both subtile WMMAs; 4 accumulators break dep chains.
    {
        const char* pa = (const char*)rir  + (ptrdiff_t)(mLo + rbase) * 4;
        const char* pb = (const char*)a_ch + (ptrdiff_t)(mLo + bbase) * 4;
        int k0 = mLo;
        for (; k0 + 4 < mHi; k0 += 8) {
            v2f a0  = *(const v2f*)(pa);          // rir[k0+rbase .. +1]
            v2f a1  = *(const v2f*)(pa + 16);     // rir[k0+4+rbase .. +1]
            f2u b00 = *(const f2u*)(pb);          // ap[k0+bbase .. +1]
            f2u b01 = *(const f2u*)(pb + 16);
            f2u b10 = *(const f2u*)(pb + 1024);   // subtile 1: +256 elements
            f2u b11 = *(const f2u*)(pb + 1040);
            v2f v00 = {b00.x, b00.y};
            v2f v01 = {b01.x, b01.y};
            v2f v10 = {b10.x, b10.y};
            v2f v11 = {b11.x, b11.y};
            acc0a = wmma4(a0, v00, acc0a);
            acc1a = wmma4(a0, v10, acc1a);
            acc0b = wmma4(a1, v01, acc0b);
            acc1b = wmma4(a1, v11, acc1b);
            pa += 32;
            pb += 32;
        }
        if (k0 < mHi) {
            v2f a0  = *(const v2f*)(pa);
            f2u b00 = *(const f2u*)(pb);
            f2u b10 = *(const f2u*)(pb + 1024);
            v2f v00 = {b00.x, b00.y};
            v2f v10 = {b10.x, b10.y};
            acc0a = wmma4(a0, v00, acc0a);
            acc1a = wmma4(a0, v10, acc1a);
        }
    }

    // Phase 3: checked epilogue
    for (int kc = mHi; kc < wHi; kc += 4)
        step_checked2(rir, a_ch, kc, rbase, bbase, acc0b, acc1b);

    // Spill wave tiles: D layout lane L -> N=(L&15), VGPR v -> M = v + 8*(L>>4)
    #pragma unroll
    for (int v = 0; v < 8; ++v) {
        int M = v + 8 * h;
        red[wave * TILE_W +       M * 16 + mq] = acc0a[v] + acc0b[v];
        red[wave * TILE_W + 256 + M * 16 + mq] = acc1a[v] + acc1b[v];
    }
    __syncthreads();

    // Cross-wave reduction: thread t owns element t of each subtile
    float s0 = 0.f, s1 = 0.f;
    #pragma unroll
    for (int w = 0; w < NWAVES; ++w) {
        s0 += red[w * TILE_W + tid];
        s1 += red[w * TILE_W + 256 + tid];
    }
    int n = n0 + tid;
    if (n < OUT_LEN)       out[ch * OUT_LEN + n]       = s0;
    if (n + 256 < OUT_LEN) out[ch * OUT_LEN + n + 256] = s1;
}

extern "C" void kernel_launch(void* const* d_in, const int* in_sizes, int n_in,
                              void* d_out, int out_size, void* d_ws, size_t ws_size,
                              hipStream_t stream) {
    (void)in_sizes; (void)n_in; (void)d_ws; (void)ws_size; (void)out_size;
    const float* audio = (const float*)d_in[0];   // (1, 4, T) float32
    const float* rir   = (const float*)d_in[1];   // (T,) float32
    float* out = (float*)d_out;                   // (1, 4, 2T-1) float32

    dim3 grid(NTILES, 4, 1);
    dim3 block(256, 1, 1);
    RoomImpulseResponse_wmma_conv<<<grid, block, 0, stream>>>(audio, rir, out);
}